// GCNModel_43473658970188
// MI455X (gfx1250) — compile-verified
//
#include <hip/hip_runtime.h>

typedef float v2f __attribute__((ext_vector_type(2)));
typedef float v8f __attribute__((ext_vector_type(8)));

#define IN_CH  128
#define HID1   64
#define HID2   32

// ---------------------------------------------------------------- utilities
__global__ void fill_zero_kernel(float* __restrict__ p, int n) {
    int i = blockIdx.x * blockDim.x + threadIdx.x;
    if (i < n) p[i] = 0.0f;
}

// deg[dst] += 1 per edge
__global__ void degree_kernel(const int* __restrict__ dst, float* __restrict__ deg, int nE) {
    int e = blockIdx.x * blockDim.x + threadIdx.x;
    if (e < nE) atomicAdd(&deg[dst[e]], 1.0f);
}

// dinv = (deg + 1)^-1/2   (in place over the deg buffer)
__global__ void dinv_kernel(float* __restrict__ dinv, int n) {
    int i = blockIdx.x * blockDim.x + threadIdx.x;
    if (i < n) dinv[i] = rsqrtf(dinv[i] + 1.0f);
}

// ---------------------------------------------------------------- WMMA GEMMs
// H[M,64] = X[M,128] @ W[128,64]; 4 waves per block, one 16x16 N-tile per wave.
__global__ __launch_bounds__(128) void gemm1_wmma(const float* __restrict__ X,
                                                  const float* __restrict__ W,
                                                  float* __restrict__ H) {
    const int lane = threadIdx.x & 31;
    const int wave = threadIdx.x >> 5;           // 0..3 -> N tile
    const int m0   = blockIdx.x * 16;
    const int n0   = wave * 16;
    const int l16  = lane & 15;
    const int khi  = (lane >> 4) * 2;            // 0 or 2
    const int row  = m0 + l16;                   // A row
    const int col  = n0 + l16;                   // B/D col
    v8f c = {};
    #pragma unroll 4
    for (int kk = 0; kk < IN_CH; kk += 4) {
        v2f a, b;
        a.x = X[row * IN_CH + kk + khi + 0];
        a.y = X[row * IN_CH + kk + khi + 1];
        b.x = W[(kk + khi + 0) * HID1 + col];
        b.y = W[(kk + khi + 1) * HID1 + col];
        c = __builtin_amdgcn_wmma_f32_16x16x4_f32(false, a, false, b,
                                                  (short)0, c, false, false);
    }
    const int mbase = m0 + ((lane >> 4) * 8);    // D: vgpr r -> M=r (lo lanes) / 8+r (hi)
    #pragma unroll
    for (int r = 0; r < 8; ++r)
        H[(mbase + r) * HID1 + col] = c[r];
}

// H2[M,32] = H1[M,64] @ W2[64,32]; 2 waves per block.
__global__ __launch_bounds__(64) void gemm2_wmma(const float* __restrict__ X,
                                                 const float* __restrict__ W,
                                                 float* __restrict__ H) {
    const int lane = threadIdx.x & 31;
    const int wave = threadIdx.x >> 5;           // 0..1 -> N tile
    const int m0   = blockIdx.x * 16;
    const int n0   = wave * 16;
    const int l16  = lane & 15;
    const int khi  = (lane >> 4) * 2;
    const int row  = m0 + l16;
    const int col  = n0 + l16;
    v8f c = {};
    #pragma unroll 4
    for (int kk = 0; kk < HID1; kk += 4) {
        v2f a, b;
        a.x = X[row * HID1 + kk + khi + 0];
        a.y = X[row * HID1 + kk + khi + 1];
        b.x = W[(kk + khi + 0) * HID2 + col];
        b.y = W[(kk + khi + 1) * HID2 + col];
        c = __builtin_amdgcn_wmma_f32_16x16x4_f32(false, a, false, b,
                                                  (short)0, c, false, false);
    }
    const int mbase = m0 + ((lane >> 4) * 8);
    #pragma unroll
    for (int r = 0; r < 8; ++r)
        H[(mbase + r) * HID2 + col] = c[r];
}

// ---------------------------------------------------------------- edge scatter
// 16 lanes per edge, float4 per lane over 64 channels.
__global__ __launch_bounds__(256) void edge_agg1_kernel(const float* __restrict__ h,
                                                        const int* __restrict__ src,
                                                        const int* __restrict__ dst,
                                                        const float* __restrict__ dinv,
                                                        float* __restrict__ agg, int nE) {
    int tid = blockIdx.x * 256 + threadIdx.x;
    int e = tid >> 4;
    if (e >= nE) return;
    int c4 = (tid & 15) * 4;
    int s = src[e], d = dst[e];
    float norm = dinv[s] * dinv[d];
    const float4 hv = *(const float4*)(h + s * HID1 + c4);
    float* ap = agg + d * HID1 + c4;
    atomicAdd(ap + 0, hv.x * norm);
    atomicAdd(ap + 1, hv.y * norm);
    atomicAdd(ap + 2, hv.z * norm);
    atomicAdd(ap + 3, hv.w * norm);
}

// 8 lanes per edge, float4 per lane over 32 channels.
__global__ __launch_bounds__(256) void edge_agg2_kernel(const float* __restrict__ h,
                                                        const int* __restrict__ src,
                                                        const int* __restrict__ dst,
                                                        const float* __restrict__ dinv,
                                                        float* __restrict__ agg, int nE) {
    int tid = blockIdx.x * 256 + threadIdx.x;
    int e = tid >> 3;
    if (e >= nE) return;
    int c4 = (tid & 7) * 4;
    int s = src[e], d = dst[e];
    float norm = dinv[s] * dinv[d];
    const float4 hv = *(const float4*)(h + s * HID2 + c4);
    float* ap = agg + d * HID2 + c4;
    atomicAdd(ap + 0, hv.x * norm);
    atomicAdd(ap + 1, hv.y * norm);
    atomicAdd(ap + 2, hv.z * norm);
    atomicAdd(ap + 3, hv.w * norm);
}

// ---------------------------------------------------------------- combines
// h1 = relu(agg1 + h1 * dinv^2 + b1)   (in place)
__global__ __launch_bounds__(256) void combine1_kernel(float* __restrict__ h,
                                                       const float* __restrict__ agg,
                                                       const float* __restrict__ dinv,
                                                       const float* __restrict__ b, int n) {
    int i = blockIdx.x * 256 + threadIdx.x;      // over n*64
    if (i >= n * HID1) return;
    int node = i >> 6, c = i & 63;
    float dd = dinv[node]; dd *= dd;
    float v = agg[i] + h[i] * dd + b[c];
    h[i] = fmaxf(v, 0.0f);
}

// per node: v[c] = relu(agg2 + h2*dinv^2 + b2);  out = sum_c v[c]*Wh[c] + bh
// one wave32 per node, shuffle reduction.
__global__ __launch_bounds__(256) void combine2_proj_kernel(const float* __restrict__ h,
                                                            const float* __restrict__ agg,
                                                            const float* __restrict__ dinv,
                                                            const float* __restrict__ b2,
                                                            const float* __restrict__ Wh,
                                                            const float* __restrict__ bh,
                                                            float* __restrict__ out, int n) {
    int lane = threadIdx.x & 31;
    int node = blockIdx.x * 8 + (threadIdx.x >> 5);
    if (node >= n) return;
    float dd = dinv[node]; dd *= dd;
    int i = node * HID2 + lane;
    float v = agg[i] + h[i] * dd + b2[lane];
    v = fmaxf(v, 0.0f);
    float acc = v * Wh[lane];
    #pragma unroll
    for (int off = 16; off > 0; off >>= 1)
        acc += __shfl_xor(acc, off, 32);
    if (lane == 0) out[node] = acc + bh[0];
}

// ---------------------------------------------------------------- launch
extern "C" void kernel_launch(void* const* d_in, const int* in_sizes, int n_in,
                              void* d_out, int out_size, void* d_ws, size_t ws_size,
                              hipStream_t stream) {
    const float* x  = (const float*)d_in[0];
    const int*   ei = (const int*)d_in[1];
    const float* W1 = (const float*)d_in[2];
    const float* b1 = (const float*)d_in[3];
    const float* W2 = (const float*)d_in[4];
    const float* b2 = (const float*)d_in[5];
    const float* Wh = (const float*)d_in[6];
    const float* bh = (const float*)d_in[7];
    float* out = (float*)d_out;

    const int N = in_sizes[0] / IN_CH;   // 50000
    const int E = in_sizes[1] / 2;       // 1,600,000
    const int* src = ei;
    const int* dst = ei + E;

    float* f    = (float*)d_ws;
    float* dinv = f;                               // N
    float* h1   = f + 65536;                       // N*64
    float* agg1 = h1 + (size_t)N * HID1;           // N*64
    float* h2   = agg1 + (size_t)N * HID1;         // N*32
    float* agg2 = h2 + (size_t)N * HID2;           // N*32

    // zero deg + agg buffers
    fill_zero_kernel<<<(N + 255) / 256, 256, 0, stream>>>(dinv, N);
    fill_zero_kernel<<<(N * HID1 + 255) / 256, 256, 0, stream>>>(agg1, N * HID1);
    fill_zero_kernel<<<(N * HID2 + 255) / 256, 256, 0, stream>>>(agg2, N * HID2);

    // degree -> dinv
    degree_kernel<<<(E + 255) / 256, 256, 0, stream>>>(dst, dinv, E);
    dinv_kernel<<<(N + 255) / 256, 256, 0, stream>>>(dinv, N);

    // layer 1
    gemm1_wmma<<<N / 16, 128, 0, stream>>>(x, W1, h1);
    edge_agg1_kernel<<<(E * 16 + 255) / 256, 256, 0, stream>>>(h1, src, dst, dinv, agg1, E);
    combine1_kernel<<<(N * HID1 + 255) / 256, 256, 0, stream>>>(h1, agg1, dinv, b1, N);

    // layer 2
    gemm2_wmma<<<N / 16, 64, 0, stream>>>(h1, W2, h2);
    edge_agg2_kernel<<<(E * 8 + 255) / 256, 256, 0, stream>>>(h2, src, dst, dinv, agg2, E);

    // combine + final projection
    combine2_proj_kernel<<<(N + 7) / 8, 256, 0, stream>>>(h2, agg2, dinv, b2, Wh, bh, out, N);
}